// OrderedGNN_678604832905
// MI455X (gfx1250) — compile-verified
//
#include <hip/hip_runtime.h>

#define NN   50000
#define EE   600000
#define INC  128
#define HH   128
#define CHN  32
#define GG   64
#define NTILES (NN / 16)          // 3125 exactly
#define TILE_BLOCKS ((NTILES + 7) / 8)

typedef float v2f __attribute__((ext_vector_type(2)));
typedef float v8f __attribute__((ext_vector_type(8)));

// ---------------------------------------------------------------------------
// Input MLP: h = LayerNorm(relu(x @ W_in + b_in))
// One wave per 16-row tile; full 16x128 output via 8 fp32 WMMA accumulators.
// ---------------------------------------------------------------------------
__global__ __launch_bounds__(256) void k_input_mlp(
    const float* __restrict__ x, const float* __restrict__ W,
    const float* __restrict__ bias, const float* __restrict__ lng,
    const float* __restrict__ lnb, float* __restrict__ h)
{
  __shared__ float Ws[INC * HH];                       // 64 KB
  for (int i = threadIdx.x; i < INC * HH; i += 256) Ws[i] = W[i];
  __syncthreads();

  const int wave = threadIdx.x >> 5;
  const int lane = threadIdx.x & 31;
  const int tile = blockIdx.x * 8 + wave;
  if (tile >= NTILES) return;
  const int row0 = tile * 16;
  const int lr   = lane & 15;
  const int half = lane >> 4;

  const v8f vzero = {0.f,0.f,0.f,0.f,0.f,0.f,0.f,0.f};
  v8f acc[8];
#pragma unroll
  for (int n0 = 0; n0 < 8; ++n0) acc[n0] = vzero;

  const float* __restrict__ xrow = x + (row0 + lr) * INC;
#pragma unroll 4
  for (int k0 = 0; k0 < INC; k0 += 4) {
    const int k = k0 + half * 2;
    v2f a; a.x = xrow[k]; a.y = xrow[k + 1];
#pragma unroll
    for (int n0 = 0; n0 < 8; ++n0) {
      v2f bf;
      bf.x = Ws[k * HH + n0 * 16 + lr];
      bf.y = Ws[(k + 1) * HH + n0 * 16 + lr];
      acc[n0] = __builtin_amdgcn_wmma_f32_16x16x4_f32(
          false, a, false, bf, (short)0, acc[n0], false, false);
    }
  }

  // bias + relu, then LayerNorm per row (row = v + 8*half, col = n0*16+lr)
  float vals[8][8];
  float p[8], q[8];
#pragma unroll
  for (int v = 0; v < 8; ++v) { p[v] = 0.f; q[v] = 0.f; }
#pragma unroll
  for (int n0 = 0; n0 < 8; ++n0) {
    const float bc = bias[n0 * 16 + lr];
#pragma unroll
    for (int v = 0; v < 8; ++v) {
      float t = acc[n0][v] + bc;
      t = t > 0.f ? t : 0.f;
      vals[n0][v] = t; p[v] += t; q[v] += t * t;
    }
  }
#pragma unroll
  for (int m = 1; m < 16; m <<= 1) {
#pragma unroll
    for (int v = 0; v < 8; ++v) {
      p[v] += __shfl_xor(p[v], m, 32);
      q[v] += __shfl_xor(q[v], m, 32);
    }
  }
  float mean[8], rstd[8];
#pragma unroll
  for (int v = 0; v < 8; ++v) {
    mean[v] = p[v] * (1.f / HH);
    float var = q[v] * (1.f / HH) - mean[v] * mean[v];
    rstd[v] = rsqrtf(var + 1e-5f);
  }
#pragma unroll
  for (int n0 = 0; n0 < 8; ++n0) {
    const int col = n0 * 16 + lr;
    const float g = lng[col], b = lnb[col];
#pragma unroll
    for (int v = 0; v < 8; ++v) {
      const int row = row0 + v + 8 * half;
      h[row * HH + col] = (vals[n0][v] - mean[v]) * rstd[v] * g + b;
    }
  }
}

// ---------------------------------------------------------------------------
// Degree init / scatter
// ---------------------------------------------------------------------------
__global__ void k_deg_init(float* __restrict__ deg) {
  int i = blockIdx.x * blockDim.x + threadIdx.x;
  if (i < NN) deg[i] = 1.f;
}

__global__ void k_deg(const int* __restrict__ src, const int* __restrict__ dst,
                      float* __restrict__ deg) {
  int e = blockIdx.x * blockDim.x + threadIdx.x;
  if (e >= EE) return;
  int s = src[e], d = dst[e];
  if (s != d) unsafeAtomicAdd(&deg[d], 1.f);
}

// ---------------------------------------------------------------------------
// Edge scatter: agg[dst] += h[src]  (one wave per edge, 4 atomics/lane)
// ---------------------------------------------------------------------------
__global__ __launch_bounds__(256) void k_agg(
    const int* __restrict__ src, const int* __restrict__ dst,
    const float* __restrict__ h, float* __restrict__ agg)
{
  const int wave = threadIdx.x >> 5;
  const int lane = threadIdx.x & 31;
  const int e = blockIdx.x * 8 + wave;
  if (e >= EE) return;
  const int s = src[e], d = dst[e];
  if (s == d) return;
  const float* __restrict__ hs = h + s * HH;
  float* __restrict__ ad = agg + d * HH;
#pragma unroll
  for (int c = 0; c < 4; ++c)
    unsafeAtomicAdd(&ad[lane + 32 * c], hs[lane + 32 * c]);
}

// ---------------------------------------------------------------------------
// Per-layer node update: gating GEMM (WMMA) + softmax + cumsum + tm merge +
// elementwise mix + LayerNorm.  h updated in place, tm updated in place.
// ---------------------------------------------------------------------------
__global__ __launch_bounds__(256) void k_layer(
    const float* __restrict__ Wtm, const float* __restrict__ btm,
    const float* __restrict__ tmg, const float* __restrict__ tmb,
    const float* __restrict__ agg, const float* __restrict__ deg,
    float* __restrict__ h, float* __restrict__ tm)
{
  __shared__ float Ws[2 * HH * CHN];                    // 32 KB
  __shared__ float raws[8][16][CHN];                    // 16 KB
  for (int i = threadIdx.x; i < 2 * HH * CHN; i += 256) Ws[i] = Wtm[i];
  __syncthreads();

  const int wave = threadIdx.x >> 5;
  const int lane = threadIdx.x & 31;
  const int tile = blockIdx.x * 8 + wave;
  if (tile >= NTILES) return;
  const int row0 = tile * 16;
  const int lr   = lane & 15;
  const int half = lane >> 4;

  const float* __restrict__ hrow = h + (row0 + lr) * HH;
  const float* __restrict__ arow = agg + (row0 + lr) * HH;
  const float invd = 1.f / deg[row0 + lr];

  const v8f vzero = {0.f,0.f,0.f,0.f,0.f,0.f,0.f,0.f};
  v8f acc[2]; acc[0] = vzero; acc[1] = vzero;

  // K = 0..127 : A from h
#pragma unroll 4
  for (int k0 = 0; k0 < HH; k0 += 4) {
    const int k = k0 + half * 2;
    v2f a; a.x = hrow[k]; a.y = hrow[k + 1];
#pragma unroll
    for (int t = 0; t < 2; ++t) {
      v2f bf;
      bf.x = Ws[k * CHN + t * 16 + lr];
      bf.y = Ws[(k + 1) * CHN + t * 16 + lr];
      acc[t] = __builtin_amdgcn_wmma_f32_16x16x4_f32(
          false, a, false, bf, (short)0, acc[t], false, false);
    }
  }
  // K = 128..255 : A from m = (agg + h) / deg
#pragma unroll 4
  for (int k0 = HH; k0 < 2 * HH; k0 += 4) {
    const int k = k0 + half * 2;
    const int c = k - HH;
    v2f a;
    a.x = (arow[c] + hrow[c]) * invd;
    a.y = (arow[c + 1] + hrow[c + 1]) * invd;
#pragma unroll
    for (int t = 0; t < 2; ++t) {
      v2f bf;
      bf.x = Ws[k * CHN + t * 16 + lr];
      bf.y = Ws[(k + 1) * CHN + t * 16 + lr];
      acc[t] = __builtin_amdgcn_wmma_f32_16x16x4_f32(
          false, a, false, bf, (short)0, acc[t], false, false);
    }
  }

  // softmax over 32 cols per row (row = v + 8*half, col = t*16 + lr)
  float z[2][8];
#pragma unroll
  for (int t = 0; t < 2; ++t) {
    const float bc = btm[t * 16 + lr];
#pragma unroll
    for (int v = 0; v < 8; ++v) z[t][v] = acc[t][v] + bc;
  }
  float mx[8];
#pragma unroll
  for (int v = 0; v < 8; ++v) mx[v] = fmaxf(z[0][v], z[1][v]);
#pragma unroll
  for (int m = 1; m < 16; m <<= 1)
#pragma unroll
    for (int v = 0; v < 8; ++v) mx[v] = fmaxf(mx[v], __shfl_xor(mx[v], m, 32));
  float sum[8];
#pragma unroll
  for (int v = 0; v < 8; ++v) {
    z[0][v] = __expf(z[0][v] - mx[v]);
    z[1][v] = __expf(z[1][v] - mx[v]);
    sum[v] = z[0][v] + z[1][v];
  }
#pragma unroll
  for (int m = 1; m < 16; m <<= 1)
#pragma unroll
    for (int v = 0; v < 8; ++v) sum[v] += __shfl_xor(sum[v], m, 32);
#pragma unroll
  for (int v = 0; v < 8; ++v) {
    const float inv = 1.f / sum[v];
    z[0][v] *= inv; z[1][v] *= inv;
  }

  // inclusive cumsum across 32 cols (scan within 16-lane halves, tile0 then tile1)
  float c0[8], c1[8];
#pragma unroll
  for (int v = 0; v < 8; ++v) { c0[v] = z[0][v]; c1[v] = z[1][v]; }
#pragma unroll
  for (int d = 1; d < 16; d <<= 1) {
#pragma unroll
    for (int v = 0; v < 8; ++v) {
      float u0 = __shfl_up(c0[v], d, 16);
      float u1 = __shfl_up(c1[v], d, 16);
      c0[v] += (lr >= d) ? u0 : 0.f;
      c1[v] += (lr >= d) ? u1 : 0.f;
    }
  }
#pragma unroll
  for (int v = 0; v < 8; ++v) c1[v] += __shfl(c0[v], 15, 16);

  // tm merge: raw = tm + (1 - tm) * cumsum ; store new tm + stash in LDS
#pragma unroll
  for (int t = 0; t < 2; ++t) {
#pragma unroll
    for (int v = 0; v < 8; ++v) {
      const int rowIn = v + 8 * half;
      const int row = row0 + rowIn;
      const int col = t * 16 + lr;
      const float told = tm[row * CHN + col];
      const float cum = t ? c1[v] : c0[v];
      const float raw = told + (1.f - told) * cum;
      tm[row * CHN + col] = raw;
      raws[wave][rowIn][col] = raw;
    }
  }

  // elementwise mix + LayerNorm (same distribution as input MLP epilogue)
  float invd2[8];
#pragma unroll
  for (int v = 0; v < 8; ++v) invd2[v] = 1.f / deg[row0 + v + 8 * half];

  float vals[8][8];
  float p[8], q[8];
#pragma unroll
  for (int v = 0; v < 8; ++v) { p[v] = 0.f; q[v] = 0.f; }
#pragma unroll
  for (int n0 = 0; n0 < 8; ++n0) {
    const int col = n0 * 16 + lr;
#pragma unroll
    for (int v = 0; v < 8; ++v) {
      const int rowIn = v + 8 * half;
      const int row = row0 + rowIn;
      const float hv = h[row * HH + col];
      const float av = agg[row * HH + col];
      const float mm = (av + hv) * invd2[v];
      const float sig = raws[wave][rowIn][col >> 2];
      const float u = hv * sig + mm * (1.f - sig);
      vals[n0][v] = u; p[v] += u; q[v] += u * u;
    }
  }
#pragma unroll
  for (int m = 1; m < 16; m <<= 1) {
#pragma unroll
    for (int v = 0; v < 8; ++v) {
      p[v] += __shfl_xor(p[v], m, 32);
      q[v] += __shfl_xor(q[v], m, 32);
    }
  }
  float mean[8], rstd[8];
#pragma unroll
  for (int v = 0; v < 8; ++v) {
    mean[v] = p[v] * (1.f / HH);
    float var = q[v] * (1.f / HH) - mean[v] * mean[v];
    rstd[v] = rsqrtf(var + 1e-5f);
  }
#pragma unroll
  for (int n0 = 0; n0 < 8; ++n0) {
    const int col = n0 * 16 + lr;
    const float g = tmg[col], b = tmb[col];
#pragma unroll
    for (int v = 0; v < 8; ++v) {
      const int row = row0 + v + 8 * half;
      h[row * HH + col] = (vals[n0][v] - mean[v]) * rstd[v] * g + b;
    }
  }
}

// ---------------------------------------------------------------------------
// Global mean pool (batch is sorted -> run-length local accumulation)
// ---------------------------------------------------------------------------
__global__ __launch_bounds__(128) void k_pool(
    const float* __restrict__ h, const int* __restrict__ batch,
    float* __restrict__ pooled, float* __restrict__ counts)
{
  const int col = threadIdx.x;           // 0..127
  const int start = blockIdx.x * 256;
  int end = start + 256; if (end > NN) end = NN;
  float acc = 0.f, cnt = 0.f; int cur = -1;
  for (int n = start; n < end; ++n) {
    const int g = batch[n];
    if (g != cur) {
      if (cur >= 0) {
        unsafeAtomicAdd(&pooled[cur * HH + col], acc);
        if (col == 0) unsafeAtomicAdd(&counts[cur], cnt);
      }
      cur = g; acc = 0.f; cnt = 0.f;
    }
    acc += h[n * HH + col];
    cnt += 1.f;
  }
  if (cur >= 0) {
    unsafeAtomicAdd(&pooled[cur * HH + col], acc);
    if (col == 0) unsafeAtomicAdd(&counts[cur], cnt);
  }
}

__global__ void k_fin(float* __restrict__ out, const float* __restrict__ counts) {
  int i = blockIdx.x * blockDim.x + threadIdx.x;
  if (i < GG * HH) out[i] /= fmaxf(counts[i >> 7], 1.f);
}

// ---------------------------------------------------------------------------
extern "C" void kernel_launch(void* const* d_in, const int* in_sizes, int n_in,
                              void* d_out, int out_size, void* d_ws, size_t ws_size,
                              hipStream_t stream) {
  const float* x    = (const float*)d_in[0];
  const int*   eidx = (const int*)d_in[1];
  const int*   batch= (const int*)d_in[2];
  const float* Win  = (const float*)d_in[3];
  const float* bin  = (const float*)d_in[4];
  const float* lng  = (const float*)d_in[5];
  const float* lnb  = (const float*)d_in[6];
  const float* Wtm  = (const float*)d_in[7];
  const float* btm  = (const float*)d_in[8];
  const float* tmg  = (const float*)d_in[9];
  const float* tmb  = (const float*)d_in[10];
  float* out = (float*)d_out;

  float* h      = (float*)d_ws;
  float* agg    = h   + (size_t)NN * HH;
  float* tm     = agg + (size_t)NN * HH;
  float* deg    = tm  + (size_t)NN * CHN;
  float* counts = deg + NN;

  const int* esrc = eidx;
  const int* edst = eidx + EE;

  hipMemsetAsync(tm, 0, (size_t)NN * CHN * sizeof(float), stream);
  k_deg_init<<<(NN + 255) / 256, 256, 0, stream>>>(deg);
  k_deg<<<(EE + 255) / 256, 256, 0, stream>>>(esrc, edst, deg);
  k_input_mlp<<<TILE_BLOCKS, 256, 0, stream>>>(x, Win, bin, lng, lnb, h);

  for (int j = 0; j < 3; ++j) {
    hipMemsetAsync(agg, 0, (size_t)NN * HH * sizeof(float), stream);
    k_agg<<<(EE + 7) / 8, 256, 0, stream>>>(esrc, edst, h, agg);
    k_layer<<<TILE_BLOCKS, 256, 0, stream>>>(Wtm, btm, tmg + j * HH, tmb + j * HH,
                                             agg, deg, h, tm);
  }

  hipMemsetAsync(out, 0, (size_t)GG * HH * sizeof(float), stream);
  hipMemsetAsync(counts, 0, GG * sizeof(float), stream);
  k_pool<<<(NN + 255) / 256, 128, 0, stream>>>(h, batch, out, counts);
  k_fin<<<(GG * HH + 255) / 256, 256, 0, stream>>>(out, counts);
}